// GELU151_39857296507280
// MI455X (gfx1250) — compile-verified
//
#include <hip/hip_runtime.h>
#include <math.h>

// ---------------------------------------------------------------------------
// Novelty-gated GELU + EMA prototype update for MI455X (gfx1250, wave32).
//
// Memory-bound workload (~512 MB of traffic vs ~1 GFLOP of matmul), so all
// math stays f32. The N x 16 similarity matmul maps exactly onto
// V_WMMA_F32_16X16X4_F32: one wave owns a 16-row tile of x and all 16 protos,
// stepping the K (=D) dimension 4 floats at a time.
// ---------------------------------------------------------------------------

typedef __attribute__((ext_vector_type(2))) float v2f;
typedef __attribute__((ext_vector_type(8))) float v8f;

#define D_DIM    4096
#define KP       16
#define N_ROWS   8192      // B*T = 4*2048
#define WPB      4         // waves per block in pass1 (128 threads)
#define P2_COLS  256
#define P2_ROWS  512
#define EPSN     1e-12f

// ---------------------------------------------------------------------------
// ws layout (floats):
//   [0]                  novelty accumulator
//   [1]                  gate scalar
//   [8 .. 8+16)          counts
//   [64 .. 64+KP*D)      segment sums
//   [64+KP*D .. +N_ROWS) assign (int)
// total ~289 KB
// ---------------------------------------------------------------------------

__global__ void init_kernel(float* __restrict__ nov, float* __restrict__ counts,
                            float* __restrict__ sums) {
    int i = blockIdx.x * 256 + threadIdx.x;
    if (i < KP * D_DIM) sums[i] = 0.0f;
    if (i < KP) counts[i] = 0.0f;
    if (i == 0) nov[0] = 0.0f;
}

// ---------------------------------------------------------------------------
// Pass 1: cosine similarity via WMMA f32 16x16x4, per-row max/argmax,
// novelty + counts accumulation.
//
// 32-bit A-matrix 16x4 layout (ISA 7.12.2): lane L holds row M = L&15,
// K-slots 2*(L>>4) + {0,1} in VGPR0/VGPR1 -> per-lane float2 load.
// We use the mirrored layout for the B fragment (protos rows), so both
// fragments use the identical addressing pattern.
// ---------------------------------------------------------------------------
__global__ void __launch_bounds__(WPB * 32)
pass1_sim_kernel(const float* __restrict__ x, const float* __restrict__ protos,
                 int* __restrict__ assign, float* __restrict__ counts,
                 float* __restrict__ nov) {
    const int lane = threadIdx.x & 31;
    const int wv   = threadIdx.x >> 5;
    const int rsel = lane & 15;   // row index (A) / proto index (B) for this lane
    const int g    = lane >> 4;   // K-slot group
    const int rowbase = (blockIdx.x * WPB + wv) * 16;

    const float* xr = x      + (size_t)(rowbase + rsel) * D_DIM + 2 * g;
    const float* pr = protos + (size_t)rsel * D_DIM + 2 * g;

    v8f c = {};
    float rnsq = 0.0f;   // partial ||x_row||^2 (this lane's K-slots)
    float pnsq = 0.0f;   // partial ||proto||^2
#pragma unroll 4
    for (int k0 = 0; k0 < D_DIM; k0 += 4) {
        v2f a = *(const v2f*)(xr + k0);
        v2f b = *(const v2f*)(pr + k0);
        rnsq += a.x * a.x + a.y * a.y;
        pnsq += b.x * b.x + b.y * b.y;
        // D = A(16x4, f32) * B(4x16, f32) + C(16x16, f32)
        c = __builtin_amdgcn_wmma_f32_16x16x4_f32(
                /*neg_a=*/false, a, /*neg_b=*/false, b,
                /*c_mod=*/(short)0, c, /*reuse_a=*/false, /*reuse_b=*/false);
    }
    // lane L covers cols == {2g, 2g+1} mod 4; lane L^16 covers the rest.
    rnsq += __shfl_xor(rnsq, 16, 32);
    pnsq += __shfl_xor(pnsq, 16, 32);
    const float rnorm = fmaxf(sqrtf(rnsq), EPSN);  // for row  (= rsel)
    const float pnorm = fmaxf(sqrtf(pnsq), EPSN);  // for proto(= rsel)

    // C layout: VGPR v, lane L -> M = v + 8*(L>>4), N = L&15.
    float novacc = 0.0f;
#pragma unroll
    for (int v = 0; v < 8; ++v) {
        // row norm for M = v + 8*g lives in lane (v + 8*g)
        float rnM = __shfl(rnorm, v + 8 * g, 32);
        float s = c[v] / (rnM * pnorm);
        float mval = s;
        int   midx = rsel;
        // max/argmax across the 16-lane group (mask < 16 stays in half-wave);
        // ties resolve to the smaller index == jnp.argmax first-max semantics.
#pragma unroll
        for (int off = 1; off < 16; off <<= 1) {
            float ov = __shfl_xor(mval, off, 32);
            int   oi = __shfl_xor(midx, off, 32);
            if (ov > mval || (ov == mval && oi < midx)) { mval = ov; midx = oi; }
        }
        if (rsel == 0) {  // lanes 0 (rows v) and 16 (rows v+8)
            assign[rowbase + v + 8 * g] = midx;
            atomicAdd(&counts[midx], 1.0f);
            novacc += 1.0f - mval;
        }
    }
    if (rsel == 0) atomicAdd(nov, novacc);
}

// ---------------------------------------------------------------------------
// Pass 2: segment sums. Each block owns a 256-col tile x 512-row chunk,
// accumulates 16x256 partials in LDS (conflict-free: addr stride 1 across
// threads), then one atomicAdd sweep. x reads are fully coalesced; assign
// reads are uniform per block -> scalar loads.
// ---------------------------------------------------------------------------
__global__ void __launch_bounds__(256)
pass2_segsum_kernel(const float* __restrict__ x, const int* __restrict__ assign,
                    float* __restrict__ sums) {
    __shared__ float acc[KP * P2_COLS];  // 16 KB
    const int t = threadIdx.x;
    for (int i = t; i < KP * P2_COLS; i += 256) acc[i] = 0.0f;
    __syncthreads();

    const int colBase = (blockIdx.x % (D_DIM / P2_COLS)) * P2_COLS;
    const int rowBase = (blockIdx.x / (D_DIM / P2_COLS)) * P2_ROWS;
    for (int r = 0; r < P2_ROWS; ++r) {
        const int row = rowBase + r;
        const int a = assign[row];  // uniform -> SGPR
        acc[a * P2_COLS + t] += x[(size_t)row * D_DIM + colBase + t];
    }
    __syncthreads();
    for (int k = 0; k < KP; ++k)
        atomicAdd(&sums[(size_t)k * D_DIM + colBase + t], acc[k * P2_COLS + t]);
}

// ---------------------------------------------------------------------------
__global__ void gate_kernel(const float* __restrict__ nov,
                            const float* __restrict__ log_alpha,
                            const float* __restrict__ log_sigma,
                            float* __restrict__ gate) {
    float novelty = nov[0] / (float)N_ROWS;
    gate[0] = 1.0f + expf(log_alpha[0]) * tanhf(expf(log_sigma[0]) * novelty);
}

// ---------------------------------------------------------------------------
// Pass 3: EMA prototype update. One block per prototype; three sweeps over D
// with two block reductions (centroid norm, then updated-vector norm).
// ---------------------------------------------------------------------------
__device__ float block_reduce_sum(float v, float* sred, float* sbc) {
    for (int off = 16; off > 0; off >>= 1) v += __shfl_down(v, off, 32);
    const int lane = threadIdx.x & 31, w = threadIdx.x >> 5;
    if (lane == 0) sred[w] = v;
    __syncthreads();
    if (threadIdx.x == 0) {
        float t = 0.0f;
        for (int i = 0; i < 8; ++i) t += sred[i];
        *sbc = t;
    }
    __syncthreads();
    return *sbc;
}

__global__ void __launch_bounds__(256)
protos_kernel(const float* __restrict__ protos, const float* __restrict__ sums,
              const float* __restrict__ counts, float* __restrict__ newp) {
    __shared__ float sred[8];
    __shared__ float sbc;
    const int k = blockIdx.x;
    const int t = threadIdx.x;
    const float cnt = counts[k];
    const float invCnt = 1.0f / fmaxf(cnt, 1.0f);

    // centroid = sums/max(cnt,1); norm1 = ||centroid||
    float ssq = 0.0f;
    for (int d = t; d < D_DIM; d += 256) {
        float cen = sums[(size_t)k * D_DIM + d] * invCnt;
        ssq += cen * cen;
    }
    const float n1 = block_reduce_sum(ssq, sred, &sbc);
    const float cScale = invCnt / fmaxf(sqrtf(n1), EPSN);

    // v = 0.95*p_old + 0.05*normalized(centroid); norm2 = ||v||
    float ssq2 = 0.0f;
    for (int d = t; d < D_DIM; d += 256) {
        float v = 0.95f * protos[(size_t)k * D_DIM + d]
                + 0.05f * (sums[(size_t)k * D_DIM + d] * cScale);
        newp[(size_t)k * D_DIM + d] = v;
        ssq2 += v * v;
    }
    const float n2 = block_reduce_sum(ssq2, sred, &sbc);
    const float vScale = 1.0f / fmaxf(sqrtf(n2), EPSN);

    const bool keep = (cnt > 0.0f);
    for (int d = t; d < D_DIM; d += 256) {
        float v = newp[(size_t)k * D_DIM + d] * vScale;
        newp[(size_t)k * D_DIM + d] = keep ? v : protos[(size_t)k * D_DIM + d];
    }
}

// ---------------------------------------------------------------------------
// Pass 4: output = gelu(x) * gate, float4 vectorized (B128 loads/stores).
// ---------------------------------------------------------------------------
__device__ inline float gelu_f(float v) {
    float v3 = v * v * v;
    return 0.5f * v * (1.0f + tanhf(0.7978845608028654f * (v + 0.044715f * v3)));
}

__global__ void __launch_bounds__(256)
gelu_kernel(const float* __restrict__ x, const float* __restrict__ gate,
            float* __restrict__ out) {
    const size_t i = (size_t)blockIdx.x * blockDim.x + threadIdx.x;
    const float g = gate[0];  // uniform -> scalar load, L2-resident
    float4 v = ((const float4*)x)[i];
    float4 r;
    r.x = gelu_f(v.x) * g;
    r.y = gelu_f(v.y) * g;
    r.z = gelu_f(v.z) * g;
    r.w = gelu_f(v.w) * g;
    ((float4*)out)[i] = r;
}

// ---------------------------------------------------------------------------
extern "C" void kernel_launch(void* const* d_in, const int* in_sizes, int n_in,
                              void* d_out, int out_size, void* d_ws, size_t ws_size,
                              hipStream_t stream) {
    const float* x         = (const float*)d_in[0];  // [4,2048,4096]
    const float* log_alpha = (const float*)d_in[1];  // scalar
    const float* log_sigma = (const float*)d_in[2];  // scalar
    const float* protos    = (const float*)d_in[3];  // [16,4096]

    float* out  = (float*)d_out;                         // output [4,2048,4096]
    float* newp = out + (size_t)N_ROWS * D_DIM;          // new_protos [16,4096]

    float* wsf    = (float*)d_ws;
    float* nov    = wsf + 0;
    float* gate   = wsf + 1;
    float* counts = wsf + 8;
    float* sums   = wsf + 64;
    int*   assign = (int*)(wsf + 64 + KP * D_DIM);

    // zero the accumulators (ws is poisoned; must be deterministic per call)
    init_kernel<<<(KP * D_DIM) / 256, 256, 0, stream>>>(nov, counts, sums);

    // similarity + argmax + novelty/counts (WMMA f32 16x16x4)
    pass1_sim_kernel<<<N_ROWS / (WPB * 16), WPB * 32, 0, stream>>>(
        x, protos, assign, counts, nov);

    // segment sums
    pass2_segsum_kernel<<<(D_DIM / P2_COLS) * (N_ROWS / P2_ROWS), 256, 0, stream>>>(
        x, assign, sums);

    // scalar gate
    gate_kernel<<<1, 1, 0, stream>>>(nov, log_alpha, log_sigma, gate);

    // EMA prototype update
    protos_kernel<<<KP, 256, 0, stream>>>(protos, sums, counts, newp);

    // gated gelu
    gelu_kernel<<<(size_t)N_ROWS * D_DIM / 4 / 256, 256, 0, stream>>>(x, gate, out);
}